// SKA_38431367364745
// MI455X (gfx1250) — compile-verified
//
#include <hip/hip_runtime.h>

#ifndef __has_builtin
#define __has_builtin(x) 0
#endif

// Problem constants (from reference): x (8,256,96,96) f32, w (8,32,9,96,96) f32
#define NN 8
#define CC 256
#define HH 96
#define WW 96
#define CW 32
#define GG 8          // groups = C / Cw
#define KK 9          // 3x3 taps
#define HW (HH * WW)  // 9216
#define TH 8          // output rows per block tile
#define LROWS (TH + 2)   // 10 LDS rows (with top/bottom halo)
#define LPITCH 104       // LDS row pitch in floats; x col j lives at LDS col j+4 (16B-aligned interior)
#define NTILES (HH / TH) // 12 row tiles

typedef int v4i __attribute__((vector_size(4 * sizeof(int))));

// ---- CDNA5 async global->LDS copy (ASYNCcnt path) --------------------------
__device__ __forceinline__ void async_copy_b128(const float* gsrc, float* lds_dst) {
#if __has_builtin(__builtin_amdgcn_global_load_async_to_lds_b128)
  __builtin_amdgcn_global_load_async_to_lds_b128(
      (__attribute__((address_space(1))) v4i*)gsrc,
      (__attribute__((address_space(3))) v4i*)lds_dst,
      0 /*imm offset*/, 0 /*cpol*/);
#else
  unsigned lds_off = (unsigned)(unsigned long long)
      (__attribute__((address_space(3))) float*)lds_dst;
  asm volatile("global_load_async_to_lds_b128 %0, %1, off"
               :: "v"(lds_off), "v"(gsrc)
               : "memory");
#endif
}

__device__ __forceinline__ void wait_async0() {
#if __has_builtin(__builtin_amdgcn_s_wait_asynccnt)
  __builtin_amdgcn_s_wait_asynccnt(0);
#else
  asm volatile("s_wait_asynccnt 0" ::: "memory");
#endif
}

// ---------------------------------------------------------------------------
__global__ __launch_bounds__(WW * TH, 1)
void SKA_involution_kernel(const float* __restrict__ x,
                           const float* __restrict__ w,
                           float* __restrict__ out) {
  __shared__ __align__(16) float tile[2][LROWS * LPITCH];

  const int tx = threadIdx.x;              // 0..95 (column)
  const int ty = threadIdx.y;              // 0..7  (row within tile)
  const int tid = ty * WW + tx;

  const int bid = blockIdx.x;              // n * (CW*NTILES) + cw * NTILES + t
  const int n   = bid / (CW * NTILES);
  const int rem = bid % (CW * NTILES);
  const int cw  = rem / NTILES;
  const int r0  = (rem % NTILES) * TH;     // first output row of this tile

  // ---- zero the halo cells once (disjoint from async-written interior, so
  // they stay zero across all 8 group iterations) ----
  if (tid < LROWS) {
    tile[0][tid * LPITCH + 3]   = 0.f;  tile[0][tid * LPITCH + 100] = 0.f;
    tile[1][tid * LPITCH + 3]   = 0.f;  tile[1][tid * LPITCH + 100] = 0.f;
  }
  if (r0 == 0 && tid < 98) {               // global row -1 is out of bounds
    tile[0][tid + 3] = 0.f;
    tile[1][tid + 3] = 0.f;
  }
  if (r0 + TH == HH && tid < 98) {         // global row 96 is out of bounds
    tile[0][(LROWS - 1) * LPITCH + tid + 3] = 0.f;
    tile[1][(LROWS - 1) * LPITCH + tid + 3] = 0.f;
  }

  // ---- per-thread async-copy assignment: 10 rows x 24 float4 groups = 240 ----
  const int lr = tid / 24;                 // LDS row 0..9      (tid < 240)
  const int cg = tid % 24;                 // 16-byte column group
  const int gr = r0 - 1 + lr;              // global x row for this LDS row
  const bool do_copy = (tid < 240) && (gr >= 0) && (gr < HH);
  const int  lds_idx = lr * LPITCH + 4 + cg * 4;           // 16B aligned
  const float* gsrc0 = x + ((long)(n * CC + cw)) * HW      // g = 0 channel
                         + (long)gr * WW + cg * 4;
  const long xstride_g = (long)CW * HW;                    // +32 channels per g

  // ---- load this pixel's 9 dynamic weights once (shared across all 8 groups)
  const int row = r0 + ty;
  const int col = tx;
  const long wbase = (((long)n * CW + cw) * KK) * HW + (long)row * WW + col;
  float wk[KK];
#pragma unroll
  for (int k = 0; k < KK; ++k)
    wk[k] = __builtin_nontemporal_load(w + wbase + (long)k * HW);

  // prefetch g = 0 into buffer 0
  if (do_copy) async_copy_b128(gsrc0, &tile[0][lds_idx]);

  const long obase = ((long)n * CC + cw) * HW + (long)row * WW + col;

#pragma unroll
  for (int g = 0; g < GG; ++g) {
    wait_async0();          // this wave's async loads (incl. current buffer) done
    __syncthreads();        // all waves' loads visible; prior readers released

    if (g + 1 < GG && do_copy)   // overlap next group's x-tile with compute
      async_copy_b128(gsrc0 + (long)(g + 1) * xstride_g,
                      &tile[(g + 1) & 1][lds_idx]);

    // 3x3 stencil: pixel (row,col) -> LDS base (row-1, col-1) = [ty][tx+3]
    const float* t = &tile[g & 1][ty * LPITCH + tx + 3];
    float acc;
    acc  = wk[0] * t[0]              + wk[1] * t[1]              + wk[2] * t[2];
    acc += wk[3] * t[LPITCH + 0]     + wk[4] * t[LPITCH + 1]     + wk[5] * t[LPITCH + 2];
    acc += wk[6] * t[2 * LPITCH + 0] + wk[7] * t[2 * LPITCH + 1] + wk[8] * t[2 * LPITCH + 2];

    __builtin_nontemporal_store(acc, out + obase + (long)g * xstride_g);
  }
}

extern "C" void kernel_launch(void* const* d_in, const int* in_sizes, int n_in,
                              void* d_out, int out_size, void* d_ws, size_t ws_size,
                              hipStream_t stream) {
  const float* x = (const float*)d_in[0];   // (8,256,96,96)
  const float* w = (const float*)d_in[1];   // (8,32,9,96,96)
  float* out = (float*)d_out;               // (8,256,96,96)
  (void)in_sizes; (void)n_in; (void)out_size; (void)d_ws; (void)ws_size;

  dim3 grid(NN * CW * NTILES);              // 8*32*12 = 3072 blocks
  dim3 block(WW, TH);                       // 96 x 8 = 768 threads = 24 wave32
  SKA_involution_kernel<<<grid, block, 0, stream>>>(x, w, out);
}